// FusedMoELayer_20358144983732
// MI455X (gfx1250) — compile-verified
//
#include <hip/hip_runtime.h>
#include <math.h>

// ---------------- problem constants ----------------
#define N_TOKENS   4096
#define NUM_IN     512
#define TOTAL_OUT  1280
#define N_EXPERTS  8
#define GATE_IN    576      // NUM_IN + TEBD_DIM
#define TEBD_DIM   64

// grouped-GEMM tiling
#define ROWS_PER_TILE 64          // token rows per row-tile (padded per expert)
#define MAX_ASSIGN   (2*N_TOKENS) // 8192 real assignments
#define MAX_PADDED   8704         // 8192 + 8*63 rounded up to 256
#define MAX_TILES    136          // bound on sum ceil(count_e/64)
#define KB           128          // K slab staged in LDS

typedef __attribute__((ext_vector_type(2))) float v2f;
typedef __attribute__((ext_vector_type(8))) float v8f;

// output is the tuple (N,512),(N,512),(N,256) concatenated flat
__device__ __forceinline__ float* out_ptr(float* out, int row, int o) {
  if (o < 512)
    return out + (size_t)row * 512 + o;
  else if (o < 1024)
    return out + (size_t)N_TOKENS * 512 + (size_t)row * 512 + (o - 512);
  else
    return out + (size_t)N_TOKENS * 1024 + (size_t)row * 256 + (o - 1024);
}

// ---------------- 1. init: zero counters, poison padding ----------------
__global__ __launch_bounds__(256) void moe_init(int* counts, int* a_ids, float* a_gate) {
  int idx = blockIdx.x * 256 + threadIdx.x;       // grid covers MAX_PADDED
  if (idx < N_EXPERTS) counts[idx] = 0;
  if (idx < MAX_PADDED) { a_ids[idx] = -1; a_gate[idx] = 0.0f; }
}

// ---------------- 2. gating: one wave per token ----------------
__global__ __launch_bounds__(256) void moe_gate(
    const float* __restrict__ x, const float* __restrict__ tebd,
    const int* __restrict__ atypes, const float* __restrict__ Wg,
    int* __restrict__ tok_expert, float* __restrict__ tok_gate,
    int* __restrict__ counts)
{
  const int lane = threadIdx.x & 31;
  const int wv   = threadIdx.x >> 5;
  const int tok  = blockIdx.x * 8 + wv;           // grid = 512 -> 4096 tokens

  const float* xr = x + (size_t)tok * NUM_IN;
  const float* tr = tebd + (size_t)atypes[tok] * TEBD_DIM;

  float acc[N_EXPERTS];
  #pragma unroll
  for (int e = 0; e < N_EXPERTS; ++e) acc[e] = 0.0f;

  for (int i = lane; i < GATE_IN; i += 32) {
    float g = (i < NUM_IN) ? xr[i] : tr[i - NUM_IN];
    const float* w = Wg + (size_t)i * N_EXPERTS;
    #pragma unroll
    for (int e = 0; e < N_EXPERTS; ++e) acc[e] += g * w[e];
  }
  #pragma unroll
  for (int off = 16; off > 0; off >>= 1) {
    #pragma unroll
    for (int e = 0; e < N_EXPERTS; ++e)
      acc[e] += __shfl_xor(acc[e], off, 32);
  }

  if (lane == 0) {
    int b0 = 0; float v0 = acc[0];
    #pragma unroll
    for (int e = 1; e < N_EXPERTS; ++e)
      if (acc[e] > v0) { v0 = acc[e]; b0 = e; }
    int b1 = -1; float v1 = -3.4e38f;
    #pragma unroll
    for (int e = 0; e < N_EXPERTS; ++e)
      if (e != b0 && acc[e] > v1) { v1 = acc[e]; b1 = e; }
    float p0 = 1.0f / (1.0f + __expf(v1 - v0));   // softmax over {v0,v1}, v0 >= v1
    float p1 = 1.0f - p0;
    tok_expert[2 * tok]     = b0;
    tok_expert[2 * tok + 1] = b1;
    tok_gate[2 * tok]       = p0;
    tok_gate[2 * tok + 1]   = p1;
    atomicAdd(&counts[b0], 1);
    atomicAdd(&counts[b1], 1);
  }
}

// ---------------- 3. scan: padded offsets + tile map ----------------
__global__ void moe_scan(const int* __restrict__ counts, int* __restrict__ cursors,
                         int* __restrict__ tile_expert, int* __restrict__ tile_base,
                         int* __restrict__ tile_count)
{
  if (threadIdx.x == 0) {
    int off = 0, t = 0;
    for (int e = 0; e < N_EXPERTS; ++e) {
      cursors[e] = off;
      int c   = counts[e];
      int pad = (c + ROWS_PER_TILE - 1) & ~(ROWS_PER_TILE - 1);
      for (int i = 0; i < pad; i += ROWS_PER_TILE) {
        tile_expert[t] = e;
        tile_base[t]   = off + i;
        ++t;
      }
      off += pad;
    }
    *tile_count = t;
  }
}

// ---------------- 4. scatter assignments into expert segments ----------------
__global__ __launch_bounds__(256) void moe_scatter(
    const int* __restrict__ tok_expert, const float* __restrict__ tok_gate,
    int* __restrict__ cursors, int* __restrict__ a_ids, float* __restrict__ a_gate)
{
  int idx = blockIdx.x * 256 + threadIdx.x;       // grid covers MAX_ASSIGN
  if (idx >= MAX_ASSIGN) return;
  int e   = tok_expert[idx];
  int pos = atomicAdd(&cursors[e], 1);
  a_ids[pos]  = idx >> 1;                          // token id
  a_gate[pos] = tok_gate[idx];
}

// ---------------- 5. bias init of full output ----------------
__global__ __launch_bounds__(256) void moe_bias(
    const int* __restrict__ tok_expert, const float* __restrict__ tok_gate,
    const float* __restrict__ bexp, float* __restrict__ out)
{
  int idx = blockIdx.x * 256 + threadIdx.x;       // grid = N_TOKENS*TOTAL_OUT/256
  int row = idx / TOTAL_OUT;
  int o   = idx - row * TOTAL_OUT;
  int e0  = tok_expert[2 * row];
  int e1  = tok_expert[2 * row + 1];
  float g0 = tok_gate[2 * row];
  float g1 = tok_gate[2 * row + 1];
  *out_ptr(out, row, o) = g0 * bexp[(size_t)e0 * TOTAL_OUT + o]
                        + g1 * bexp[(size_t)e1 * TOTAL_OUT + o];
}

// ---------------- 6. grouped GEMM with f32 WMMA ----------------
// grid: (MAX_TILES, TOTAL_OUT/256), block: 128 threads (4 waves)
// wave w covers cols [blockIdx.y*256 + 64w, +64): 4 row-tiles x 4 col-tiles of 16x16
// Every B fragment (2 loads) feeds 4 WMMAs -> 0.5 global loads per WMMA.
__global__ __launch_bounds__(128) void moe_gemm(
    const float* __restrict__ x, const float* __restrict__ Wexp,
    const int* __restrict__ tile_expert, const int* __restrict__ tile_base,
    const int* __restrict__ tile_count,
    const int* __restrict__ a_ids, const float* __restrict__ a_gate,
    float* __restrict__ out)
{
  if ((int)blockIdx.x >= *tile_count) return;

  __shared__ float As[ROWS_PER_TILE][KB];    // 64 x 128 f32 = 32 KB
  __shared__ float gate_s[ROWS_PER_TILE];
  __shared__ int   ids_s[ROWS_PER_TILE];

  const int e    = tile_expert[blockIdx.x];
  const int base = tile_base[blockIdx.x];
  const int tid  = threadIdx.x;
  const int lane = tid & 31;
  const int wv   = tid >> 5;

  if (tid < ROWS_PER_TILE) {
    ids_s[tid]  = a_ids[base + tid];
    gate_s[tid] = a_gate[base + tid];
  }
  __syncthreads();

  const float* We = Wexp + (size_t)e * (NUM_IN * TOTAL_OUT);
  const int n  = lane & 15;        // output column within 16x16 tile
  const int kh = lane >> 4;        // which K pair of the 16x16x4 fragment
  const int m  = lane & 15;        // A row within tile-half
  const int cb = blockIdx.y * 256 + wv * 64;

  v8f acc[4][4];
  #pragma unroll
  for (int rt = 0; rt < 4; ++rt)
    #pragma unroll
    for (int c = 0; c < 4; ++c)
      #pragma unroll
      for (int v = 0; v < 8; ++v) acc[rt][c][v] = 0.0f;

  for (int s = 0; s < NUM_IN / KB; ++s) {
    // --- stage gathered A slab: 64 rows x KB cols (2 threads per row) ---
    {
      int r = tid >> 1;
      int q = tid & 1;
      int tok = ids_s[r];
      const float4* src = (const float4*)(x + (size_t)(tok < 0 ? 0 : tok) * NUM_IN + s * KB);
      float4* dst = (float4*)&As[r][0];
      #pragma unroll
      for (int j = 0; j < KB / 8; ++j) {     // 16 float4 per thread
        int c4 = q + 2 * j;
        float4 v = (tok >= 0) ? src[c4] : make_float4(0.f, 0.f, 0.f, 0.f);
        dst[c4] = v;
      }
    }
    __syncthreads();

    for (int kk = 0; kk < KB / 4; ++kk) {
      const int kl = 4 * kk + 2 * kh;                 // this lane's K pair (local)
      v2f a0 = *(const v2f*)&As[m][kl];               // rows  0..15
      v2f a1 = *(const v2f*)&As[m + 16][kl];          // rows 16..31
      v2f a2 = *(const v2f*)&As[m + 32][kl];          // rows 32..47
      v2f a3 = *(const v2f*)&As[m + 48][kl];          // rows 48..63
      const int gk = s * KB + kl;                     // global K
      const float* bp = We + (size_t)gk * TOTAL_OUT + cb + n;
      #pragma unroll
      for (int c = 0; c < 4; ++c) {
        v2f b;
        b.x = bp[16 * c];
        b.y = bp[TOTAL_OUT + 16 * c];
        acc[0][c] = __builtin_amdgcn_wmma_f32_16x16x4_f32(
            false, a0, false, b, (short)0, acc[0][c], false, false);
        acc[1][c] = __builtin_amdgcn_wmma_f32_16x16x4_f32(
            false, a1, false, b, (short)0, acc[1][c], false, false);
        acc[2][c] = __builtin_amdgcn_wmma_f32_16x16x4_f32(
            false, a2, false, b, (short)0, acc[2][c], false, false);
        acc[3][c] = __builtin_amdgcn_wmma_f32_16x16x4_f32(
            false, a3, false, b, (short)0, acc[3][c], false, false);
      }
    }
    __syncthreads();
  }

  // --- epilogue: gate-scale + atomic accumulate into split output ---
  #pragma unroll
  for (int rt = 0; rt < 4; ++rt) {
    #pragma unroll
    for (int v = 0; v < 8; ++v) {
      int mrow = rt * 16 + v + (kh ? 8 : 0);          // C layout: m = vgpr + 8*(lane/16)
      int tok = ids_s[mrow];
      if (tok < 0) continue;
      float g = gate_s[mrow];
      #pragma unroll
      for (int c = 0; c < 4; ++c) {
        int o = cb + 16 * c + n;
        atomicAdd(out_ptr(out, tok, o), g * acc[rt][c][v]);
      }
    }
  }
}

// ---------------- launch ----------------
extern "C" void kernel_launch(void* const* d_in, const int* in_sizes, int n_in,
                              void* d_out, int out_size, void* d_ws, size_t ws_size,
                              hipStream_t stream) {
  const float* x     = (const float*)d_in[0];
  const float* tebd  = (const float*)d_in[1];
  const int*   atyp  = (const int*)  d_in[2];
  const float* Wg    = (const float*)d_in[3];
  const float* Wexp  = (const float*)d_in[4];
  const float* bexp  = (const float*)d_in[5];
  float* out = (float*)d_out;

  char* ws = (char*)d_ws;
  int*   counts      = (int*)(ws + 0);                       // 8
  int*   cursors     = (int*)(ws + 32);                      // 8
  int*   tile_count  = (int*)(ws + 64);                      // 1
  int*   tile_expert = (int*)(ws + 128);                     // <=136 (pad 272)
  int*   tile_base   = (int*)(ws + 128 + 1088);              // <=136
  int*   tok_expert  = (int*)(ws + 2304);                    // 8192 ints
  float* tok_gate    = (float*)(ws + 2304 + 32768);          // 8192 floats
  int*   a_ids       = (int*)(ws + 2304 + 65536);            // 8704 ints
  float* a_gate      = (float*)(ws + 2304 + 65536 + 34816);  // 8704 floats

  moe_init<<<MAX_PADDED / 256, 256, 0, stream>>>(counts, a_ids, a_gate);
  moe_gate<<<N_TOKENS / 8, 256, 0, stream>>>(x, tebd, atyp, Wg, tok_expert, tok_gate, counts);
  moe_scan<<<1, 32, 0, stream>>>(counts, cursors, tile_expert, tile_base, tile_count);
  moe_scatter<<<MAX_ASSIGN / 256, 256, 0, stream>>>(tok_expert, tok_gate, cursors, a_ids, a_gate);
  moe_bias<<<(N_TOKENS * TOTAL_OUT) / 256, 256, 0, stream>>>(tok_expert, tok_gate, bexp, out);
  moe_gemm<<<dim3(MAX_TILES, TOTAL_OUT / 256), 128, 0, stream>>>(
      x, Wexp, tile_expert, tile_base, tile_count, a_ids, a_gate, out);
}